// GraphSAGE_31662498906633
// MI455X (gfx1250) — compile-verified
//
#include <hip/hip_runtime.h>
#include <hip/hip_bf16.h>

// ---------------------------------------------------------------------------
// GraphSAGE forward for MI455X (gfx1250, wave32, WMMA).
// GEMMs use V_WMMA_F32_16X16X4_F32; weights staged into LDS via the CDNA5
// async-to-LDS path (GLOBAL_LOAD_ASYNC_TO_LDS_B128 + s_wait_asynccnt) when
// the toolchain exposes it; scatter phase uses global float atomics
// (memory-bound, ~4.8 GB @ 23.3 TB/s).
// Fusions: inv_deg row-scale folded into GEMM A-fragment loads (pass 0);
// embeddings copy folded into the final BatchNorm-apply.
// ---------------------------------------------------------------------------

typedef __attribute__((ext_vector_type(2))) float v2f;
typedef __attribute__((ext_vector_type(8))) float v8f;

// Types for the async-to-LDS builtin (per hipcc diagnostic: int __vector(4)
// pointers; global source in addrspace(1), LDS destination in addrspace(3)).
typedef int v4i_vec __attribute__((vector_size(16)));
typedef __attribute__((address_space(1))) v4i_vec* g_v4i_ptr;
typedef __attribute__((address_space(3))) v4i_vec* l_v4i_ptr;

#define GN 100000
#define GE 1600000
#define GIN 128
#define GH 256
#define GL 3
#define GC 2

#if __has_builtin(__builtin_amdgcn_global_load_async_to_lds_b128)
#define USE_ASYNC_LDS 1
#else
#define USE_ASYNC_LDS 0
#endif

// ---------------------------------------------------------------------------
// Utility kernels
// ---------------------------------------------------------------------------
__global__ void k_zero(float* __restrict__ p, long n) {
    long i = (long)blockIdx.x * blockDim.x + threadIdx.x;
    if (i < n) p[i] = 0.0f;
}

__global__ void k_zero4(float* __restrict__ p, long n4) {  // n4 = count of float4
    long i = (long)blockIdx.x * blockDim.x + threadIdx.x;
    if (i < n4) ((float4*)p)[i] = make_float4(0.f, 0.f, 0.f, 0.f);
}

__global__ void k_degree(const int* __restrict__ dst, float* __restrict__ deg, int E) {
    int e = blockIdx.x * blockDim.x + threadIdx.x;
    if (e < E) atomicAdd(&deg[dst[e]], 1.0f);
}

__global__ void k_invdeg(const float* __restrict__ deg, float* __restrict__ invd, int n) {
    int i = blockIdx.x * blockDim.x + threadIdx.x;
    if (i < n) invd[i] = 1.0f / fmaxf(deg[i], 1.0f);
}

// ---------------------------------------------------------------------------
// Edge scatter: agg[dst] += h[src], float4 granularity (64 quads per row).
// ---------------------------------------------------------------------------
__global__ void k_scatter(const float* __restrict__ h, const int* __restrict__ src,
                          const int* __restrict__ dst, float* __restrict__ agg, int E) {
    long idx = (long)blockIdx.x * blockDim.x + threadIdx.x;
    long total = (long)E * (GH / 4);
    if (idx >= total) return;
    int e = (int)(idx >> 6);      // GH/4 == 64
    int q = (int)(idx & 63);
    int s = src[e], d = dst[e];
    const float4 v = *(const float4*)&h[(size_t)s * GH + q * 4];
    float* p = &agg[(size_t)d * GH + q * 4];
    atomicAdd(p + 0, v.x);
    atomicAdd(p + 1, v.y);
    atomicAdd(p + 2, v.z);
    atomicAdd(p + 3, v.w);
}

// ---------------------------------------------------------------------------
// WMMA GEMM: out[M x NC] = act( diag(rowscale)*A @ W (+ A2@W2) + bias )
// Block: 256 threads = 8 waves; block tile = 64 rows x NC cols.
// Wave (rgrp 0..3, cgrp 0..1): 16 rows x NC/2 cols -> NC/32 accum tiles (v8f).
// W staged in LDS in K-chunks of 32; rowscale applies to pass-0 A only.
// ---------------------------------------------------------------------------
template <int KTOT, int NC, bool RELU>
__global__ __launch_bounds__(256, 2)
void k_wmma_gemm(const float* __restrict__ A, const float* __restrict__ W,
                 const float* __restrict__ A2, const float* __restrict__ W2,
                 const float* __restrict__ rowscale, const float* __restrict__ bias,
                 float* __restrict__ out, int M) {
    constexpr int KC = 32;
    constexpr int NT = NC / 32;                    // accum tiles per wave
    __shared__ __align__(16) float lw[KC * NC];    // 32KB (NC=256) / 16KB (NC=128)

    const int tid  = threadIdx.x;
    const int wave = tid >> 5;
    const int lane = tid & 31;
    const int rgrp = wave >> 1;
    const int cgrp = wave & 1;
    const int rbase = blockIdx.x * 64 + rgrp * 16;
    const int cbase = cgrp * (NC / 2);
    const int arow  = rbase + (lane & 15);
    const bool rok  = (arow < M);
    const int koff  = (lane >> 4) * 2;             // lanes 0-15: K{0,1}; 16-31: K{2,3}

    v8f acc[NT] = {};

    const int npass = (A2 != nullptr) ? 2 : 1;
    for (int pass = 0; pass < npass; ++pass) {
        const float* __restrict__ Ap = pass ? A2 : A;
        const float* __restrict__ Wp = pass ? W2 : W;
        // inv-deg (mean aggregation) fused as a row scale on the pass-0 A.
        const float sc = (pass == 0 && rowscale != nullptr && rok) ? rowscale[arow] : 1.0f;
        for (int kc = 0; kc < KTOT; kc += KC) {
            __syncthreads();
            // Cooperative stage of W[kc..kc+31, 0..NC) into LDS, 16B per thread.
#if USE_ASYNC_LDS
#pragma unroll
            for (int i = 0; i < (KC * NC) / (256 * 4); ++i) {
                int e4 = (i * 256 + tid) * 4;
                int k = e4 / NC;
                int n = e4 % NC;
                // Direct memory->LDS DMA, tracked by ASYNCcnt (no VGPR round trip).
                __builtin_amdgcn_global_load_async_to_lds_b128(
                    (g_v4i_ptr)&Wp[(size_t)(kc + k) * NC + n],
                    (l_v4i_ptr)&lw[k * NC + n],
                    /*offset=*/0, /*cpol=*/0);
            }
#if __has_builtin(__builtin_amdgcn_s_wait_asynccnt)
            __builtin_amdgcn_s_wait_asynccnt(0);
#else
            asm volatile("s_wait_asynccnt 0x0" ::: "memory");
#endif
#else
#pragma unroll
            for (int i = 0; i < (KC * NC) / (256 * 4); ++i) {
                int e4 = (i * 256 + tid) * 4;
                int k = e4 / NC;
                int n = e4 % NC;
                *(float4*)&lw[k * NC + n] = *(const float4*)&Wp[(size_t)(kc + k) * NC + n];
            }
#endif
            __syncthreads();

            if (rok && kc + KC < KTOT)
                __builtin_prefetch(&Ap[(size_t)arow * KTOT + kc + KC], 0, 0);

#pragma unroll
            for (int ks = 0; ks < KC / 4; ++ks) {
                // A fragment (16x4 fp32 ISA layout), 8-byte global load, L2-hot.
                int ka = kc + ks * 4 + koff;
                v2f a = {0.0f, 0.0f};
                if (rok) a = *(const v2f*)&Ap[(size_t)arow * KTOT + ka];
                a.x *= sc;
                a.y *= sc;
                int kb = ks * 4 + koff;
#pragma unroll
                for (int t = 0; t < NT; ++t) {
                    int col = cbase + t * 16 + (lane & 15);
                    v2f b;
                    b.x = lw[kb * NC + col];
                    b.y = lw[(kb + 1) * NC + col];
                    acc[t] = __builtin_amdgcn_wmma_f32_16x16x4_f32(
                        /*neg_a=*/false, a, /*neg_b=*/false, b,
                        /*c_mod=*/(short)0, acc[t],
                        /*reuse_a=*/false, /*reuse_b=*/false);
                }
            }
        }
    }

    // Epilogue: bias (+ReLU), store per documented C/D layout
    // (VGPR r: lanes 0-15 -> M=r, lanes 16-31 -> M=r+8; N = lane&15).
#pragma unroll
    for (int t = 0; t < NT; ++t) {
        int col = cbase + t * 16 + (lane & 15);
        float bv = bias ? bias[col] : 0.0f;
#pragma unroll
        for (int r = 0; r < 8; ++r) {
            int row = rbase + r + ((lane >> 4) << 3);
            float v = acc[t][r] + bv;
            if (RELU) v = fmaxf(v, 0.0f);
            if (row < M) out[(size_t)row * NC + col] = v;
        }
    }
}

// ---------------------------------------------------------------------------
// BatchNorm: per-channel partial sums (one channel per thread, row slab per
// block, coalesced across the 256 channels), then fused apply+ReLU+residual.
// ---------------------------------------------------------------------------
__global__ void k_bn_stats(const float* __restrict__ o, float* __restrict__ sum,
                           float* __restrict__ sq, int M) {
    int c = threadIdx.x;  // 256 channels
    int r0 = blockIdx.x * 256;
    int rend = r0 + 256 < M ? r0 + 256 : M;
    float s = 0.0f, q = 0.0f;
    for (int r = r0; r < rend; ++r) {
        float v = o[(size_t)r * GH + c];
        s += v;
        q += v * v;
    }
    atomicAdd(&sum[c], s);
    atomicAdd(&sq[c], q);
}

__global__ void k_bn_apply(const float* __restrict__ o, float* __restrict__ h,
                           const float* __restrict__ sum, const float* __restrict__ sq,
                           const float* __restrict__ g, const float* __restrict__ b,
                           int M, int residual, float* __restrict__ emb2) {
    long idx = (long)blockIdx.x * blockDim.x + threadIdx.x;
    long total = (long)M * GH;
    if (idx >= total) return;
    int c = (int)(idx & (GH - 1));
    float mu  = sum[c] * (1.0f / (float)GN);
    float var = sq[c] * (1.0f / (float)GN) - mu * mu;
    float v = (o[idx] - mu) * rsqrtf(var + 1e-5f) * g[c] + b[c];
    v = fmaxf(v, 0.0f);
    if (residual) v += h[idx];
    h[idx] = v;
    if (emb2) emb2[idx] = v;  // final layer: embeddings output, fused (no copy pass)
}

// ---------------------------------------------------------------------------
// fc2: [N,128] @ [128,2] + bias -> logits; trivially VALU-bound, scalar kernel.
// ---------------------------------------------------------------------------
__global__ void k_fc2(const float* __restrict__ z, const float* __restrict__ w,
                      const float* __restrict__ b, float* __restrict__ logits, int M) {
    int r = blockIdx.x * blockDim.x + threadIdx.x;
    if (r >= M) return;
    float a0 = b[0], a1 = b[1];
    const float* zr = z + (size_t)r * (GH / 2);
#pragma unroll 4
    for (int k = 0; k < GH / 2; ++k) {
        float zv = zr[k];
        a0 += zv * w[k * 2 + 0];
        a1 += zv * w[k * 2 + 1];
    }
    logits[(size_t)r * 2 + 0] = a0;
    logits[(size_t)r * 2 + 1] = a1;
}

// ---------------------------------------------------------------------------
// Host orchestration
// ---------------------------------------------------------------------------
extern "C" void kernel_launch(void* const* d_in, const int* in_sizes, int n_in,
                              void* d_out, int out_size, void* d_ws, size_t ws_size,
                              hipStream_t stream) {
    const float* x       = (const float*)d_in[0];
    const int*   ei      = (const int*)d_in[1];       // [2,E]: src row 0, dst row 1
    const float* proj_w  = (const float*)d_in[2];
    const float* proj_b  = (const float*)d_in[3];
    const float* lin_l_w = (const float*)d_in[4];     // [L,H,H]
    const float* lin_l_b = (const float*)d_in[5];     // [L,H]
    const float* lin_r_w = (const float*)d_in[6];     // [L,H,H]
    const float* bn_g    = (const float*)d_in[7];     // [L,H]
    const float* bn_b    = (const float*)d_in[8];     // [L,H]
    const float* fc1_w   = (const float*)d_in[9];     // [H,H/2]
    const float* fc1_b   = (const float*)d_in[10];    // [H/2]
    const float* fc2_w   = (const float*)d_in[11];    // [H/2,C]
    const float* fc2_b   = (const float*)d_in[12];    // [C]

    const int* src = ei;
    const int* dst = ei + GE;

    // Workspace carve-up (fp32): h, agg, outb are N*H each; z aliases agg.
    float* ws   = (float*)d_ws;
    float* h    = ws;
    float* agg  = h + (size_t)GN * GH;
    float* outb = agg + (size_t)GN * GH;
    float* deg  = outb + (size_t)GN * GH;
    float* invd = deg + GN;
    float* bsum = invd + GN;
    float* bsq  = bsum + GH;
    float* z    = agg;  // reuse after message-passing layers finish

    float* logits = (float*)d_out;
    float* emb    = logits + (size_t)GN * GC;

    const int T = 256;
    const long NH  = (long)GN * GH;
    const long NH4 = NH / 4;

    // ---- degrees ----
    k_zero<<<(GN + T - 1) / T, T, 0, stream>>>(deg, GN);
    k_degree<<<(GE + T - 1) / T, T, 0, stream>>>(dst, deg, GE);
    k_invdeg<<<(GN + T - 1) / T, T, 0, stream>>>(deg, invd, GN);

    // ---- h = relu(x @ proj_w + proj_b) ----
    k_wmma_gemm<GIN, GH, true><<<(GN + 63) / 64, T, 0, stream>>>(
        x, proj_w, nullptr, nullptr, nullptr, proj_b, h, GN);

    // ---- SAGE layers ----
    for (int i = 0; i < GL; ++i) {
        k_zero4<<<(int)((NH4 + T - 1) / T), T, 0, stream>>>(agg, NH4);
        long sc_items = (long)GE * (GH / 4);
        k_scatter<<<(int)((sc_items + T - 1) / T), T, 0, stream>>>(h, src, dst, agg, GE);

        // out = (agg * inv_deg) @ Wl + lin_l_b + h @ Wr  (dual-pass WMMA GEMM,
        // mean-aggregation scale fused into pass-0 A-fragment loads)
        k_wmma_gemm<GH, GH, false><<<(GN + 63) / 64, T, 0, stream>>>(
            agg, lin_l_w + (size_t)i * GH * GH,
            h,   lin_r_w + (size_t)i * GH * GH,
            invd, lin_l_b + (size_t)i * GH, outb, GN);

        // BatchNorm (batch stats) + ReLU + residual, h updated in place.
        // Last layer also streams embeddings straight to d_out (fused copy).
        k_zero<<<1, 2 * GH, 0, stream>>>(bsum, 2 * GH);  // bsum|bsq contiguous
        k_bn_stats<<<(GN + 255) / 256, 256, 0, stream>>>(outb, bsum, bsq, GN);
        k_bn_apply<<<(int)((NH + T - 1) / T), T, 0, stream>>>(
            outb, h, bsum, bsq, bn_g + (size_t)i * GH, bn_b + (size_t)i * GH, GN,
            i > 0, (i == GL - 1) ? emb : nullptr);
    }

    // ---- z = relu(h @ fc1_w + fc1_b) ----
    k_wmma_gemm<GH, GH / 2, true><<<(GN + 63) / 64, T, 0, stream>>>(
        h, fc1_w, nullptr, nullptr, nullptr, fc1_b, z, GN);

    // ---- logits = z @ fc2_w + fc2_b ----
    k_fc2<<<(GN + T - 1) / T, T, 0, stream>>>(z, fc2_w, fc2_b, logits, GN);
}